// GuidedFilter_11029476016224
// MI455X (gfx1250) — compile-verified
//
#include <hip/hip_runtime.h>

#define PDIM 2048
#define PSIZE ((size_t)2048 * 2048)

typedef __attribute__((ext_vector_type(2))) float v2f;
typedef __attribute__((ext_vector_type(8))) float v8f;

// Band matrix value: 1.0 if out-index window [i, i+16] covers input index k.
__device__ __forceinline__ float band(int i, int k) {
    return (k >= i && k <= i + 16) ? 1.0f : 0.0f;
}

__device__ __forceinline__ float winlen(int i) {
    return (float)(min(PDIM - 1, i + 8) - max(0, i - 8) + 1);
}

// ---------------- Stage 1: channel reduction (float4 streaming) ------------
// ws planes: [0]=SX  [1]=SY  [2]=SXY  [3]=SXX   (all unscaled)
__global__ void reduce_kernel(const float* __restrict__ x,
                              const float* __restrict__ y,
                              float* __restrict__ ws) {
    size_t q = (size_t)blockIdx.x * blockDim.x + threadIdx.x;   // quad index
    if (q >= PSIZE / 4) return;
    const float4* x4 = (const float4*)x;
    const float4* y4 = (const float4*)y;
    float4* w4 = (float4*)ws;
    const size_t PQ = PSIZE / 4;

    float4 a0 = x4[q], a1 = x4[q + PQ], a2 = x4[q + 2 * PQ], a3 = x4[q + 3 * PQ];
    float4 b0 = y4[q], b1 = y4[q + PQ], b2 = y4[q + 2 * PQ], b3 = y4[q + 3 * PQ];

    float4 sx, sy, sxy, sxx;
    sx.x = a0.x + a1.x + a2.x + a3.x;  sx.y = a0.y + a1.y + a2.y + a3.y;
    sx.z = a0.z + a1.z + a2.z + a3.z;  sx.w = a0.w + a1.w + a2.w + a3.w;
    sy.x = b0.x + b1.x + b2.x + b3.x;  sy.y = b0.y + b1.y + b2.y + b3.y;
    sy.z = b0.z + b1.z + b2.z + b3.z;  sy.w = b0.w + b1.w + b2.w + b3.w;
    sxy.x = a0.x*b0.x + a1.x*b1.x + a2.x*b2.x + a3.x*b3.x;
    sxy.y = a0.y*b0.y + a1.y*b1.y + a2.y*b2.y + a3.y*b3.y;
    sxy.z = a0.z*b0.z + a1.z*b1.z + a2.z*b2.z + a3.z*b3.z;
    sxy.w = a0.w*b0.w + a1.w*b1.w + a2.w*b2.w + a3.w*b3.w;
    sxx.x = a0.x*a0.x + a1.x*a1.x + a2.x*a2.x + a3.x*a3.x;
    sxx.y = a0.y*a0.y + a1.y*a1.y + a2.y*a2.y + a3.y*a3.y;
    sxx.z = a0.z*a0.z + a1.z*a1.z + a2.z*a2.z + a3.z*a3.z;
    sxx.w = a0.w*a0.w + a1.w*a1.w + a2.w*a2.w + a3.w*a3.w;

    w4[q] = sx;  w4[q + PQ] = sy;  w4[q + 2 * PQ] = sxy;  w4[q + 3 * PQ] = sxx;
}

// ---------------- Stage 2/4: separable 17x17 box filter via WMMA -----------
// One workgroup (8 waves) produces a 64x64 output tile of one plane.
// Each wave computes two 16x16 sub-tiles:
//   H(32x16)   = patch(32x32) * W2(32x16)   [16 WMMAs]
//   out(16x16) = W(16x32) * H(32x16)        [8 WMMAs]
// W/W2 are constant 0/1 band matrices; the per-lane fragment values are
// identical for horizontal-B and vertical-A, precomputed once per thread.
__global__ __launch_bounds__(256) void box_kernel(const float* __restrict__ srcBase,
                                                  float* __restrict__ dstBase) {
    const float* src = srcBase + (size_t)blockIdx.z * PSIZE;
    float* dst       = dstBase + (size_t)blockIdx.z * PSIZE;
    const int I0 = blockIdx.y * 64;
    const int J0 = blockIdx.x * 64;

    __shared__ float sp[80][84];      // 80x80 padded patch (halo = 8), stride 84
    __shared__ float hs[8][32][17];   // per-wave H scratch (32 rows x 16 cols, pad 17)

    // Cooperative coalesced patch load with zero padding (6400 = 25 * 256 exactly).
    for (int pos = threadIdx.x; pos < 80 * 80; pos += 256) {
        int r = pos / 80;
        int c = pos - r * 80;
        int gi = I0 - 8 + r;
        int gj = J0 - 8 + c;
        float v = 0.0f;
        if (gi >= 0 && gi < PDIM && gj >= 0 && gj < PDIM)
            v = src[(size_t)gi * PDIM + gj];
        sp[r][c] = v;
    }
    __syncthreads();

    const int wave = threadIdx.x >> 5;
    const int lane = threadIdx.x & 31;
    const int m    = lane & 15;   // A row / B col / D col index
    const int half = lane >> 4;   // lane-half select

    // Hoisted constant band fragments: cw[s] holds values for K = 4s+2*half {+0,+1}.
    v2f cw[8];
#pragma unroll
    for (int s = 0; s < 8; ++s) {
        const int kb = s * 4 + half * 2;
        cw[s].x = band(m, kb);
        cw[s].y = band(m, kb + 1);
    }

    for (int sidx = 0; sidx < 2; ++sidx) {
        const int st = wave + sidx * 8;   // sub-tile 0..15
        const int ti = st >> 2;
        const int tj = st & 3;
        const int pr = ti * 16;           // patch row base of this sub-tile
        const int pc = tj * 16;           // patch col base

        v8f accT = {};                    // H rows 0..15
        v8f accB = {};                    // H rows 16..31
#pragma unroll
        for (int s = 0; s < 8; ++s) {
            const int kb = s * 4 + half * 2;  // A: K = v + 2*half within 4-window
            v2f aT = { sp[pr + m][pc + kb],      sp[pr + m][pc + kb + 1] };
            v2f aB = { sp[pr + 16 + m][pc + kb], sp[pr + 16 + m][pc + kb + 1] };
            accT = __builtin_amdgcn_wmma_f32_16x16x4_f32(false, aT, false, cw[s], (short)0, accT, false, false);
            accB = __builtin_amdgcn_wmma_f32_16x16x4_f32(false, aB, false, cw[s], (short)0, accB, false, false);
        }

        // Spill H (D layout: row = v + 8*half, col = m) to per-wave LDS scratch.
#pragma unroll
        for (int v = 0; v < 8; ++v) {
            hs[wave][v + 8 * half][m]      = accT[v];
            hs[wave][16 + v + 8 * half][m] = accB[v];
        }

        // Vertical pass: out = W(16x32) * H(32x16).
        v8f accO = {};
#pragma unroll
        for (int s = 0; s < 8; ++s) {
            const int kb = s * 4 + half * 2;
            v2f bH = { hs[wave][kb][m], hs[wave][kb + 1][m] };
            accO = __builtin_amdgcn_wmma_f32_16x16x4_f32(false, cw[s], false, bH, (short)0, accO, false, false);
        }

        // Write out (D layout): row = v + 8*half, col = m.
#pragma unroll
        for (int v = 0; v < 8; ++v) {
            const int gi = I0 + pr + v + 8 * half;
            const int gj = J0 + pc + m;
            dst[(size_t)gi * PDIM + gj] = accO[v];
        }
    }
}

// ---------------- Stage 3: per-pixel guided-filter coefficients ------------
__global__ void coeff_kernel(const float* __restrict__ box,  // planes: BX,BY,BXY,BXX
                             float* __restrict__ Aout,
                             float* __restrict__ bOut,
                             float* __restrict__ meanYOut) {
    size_t q = (size_t)blockIdx.x * blockDim.x + threadIdx.x;
    if (q >= PSIZE / 4) return;
    const size_t PQ = PSIZE / 4;
    const float4* b4 = (const float4*)box;
    size_t idx = q * 4;
    int i  = (int)(idx >> 11);
    int j0 = (int)(idx & 2047);
    float wh = winlen(i);

    float4 bx = b4[q], by = b4[q + PQ], bxy = b4[q + 2 * PQ], bxx = b4[q + 3 * PQ];
    float4 A4, B4;
    float* pbx = &bx.x; float* pby = &by.x; float* pbxy = &bxy.x; float* pbxx = &bxx.x;
    float* pA = &A4.x;  float* pB = &B4.x;
#pragma unroll
    for (int c = 0; c < 4; ++c) {
        float n   = 4.0f * wh * winlen(j0 + c);
        float mx  = pbx[c] / n;
        float my  = pby[c] / n;
        float cov = pbxy[c] / n - mx * my;
        float var = pbxx[c] / n - mx * mx;
        float A   = cov / (var + 0.01f);   // EPS(1e-8) / scale^2(1e-6) = 0.01
        pA[c] = A;
        pB[c] = my - A * mx;
    }
    ((float4*)Aout)[q]     = A4;
    ((float4*)bOut)[q]     = B4;
    ((float4*)meanYOut)[q] = by;           // mean_y_tmp = box_filter(y) (unscaled)
}

// ---------------- Stage 5: final output ------------------------------------
__global__ void final_kernel(const float* __restrict__ box,  // planes: boxA, boxB
                             const float* __restrict__ x2,   // original x channel 2
                             float* __restrict__ out) {
    size_t q = (size_t)blockIdx.x * blockDim.x + threadIdx.x;
    if (q >= PSIZE / 4) return;
    const size_t PQ = PSIZE / 4;
    size_t idx = q * 4;
    int i  = (int)(idx >> 11);
    int j0 = (int)(idx & 2047);
    float wh = winlen(i);

    float4 ba = ((const float4*)box)[q];
    float4 bb = ((const float4*)box)[q + PQ];
    float4 xv = ((const float4*)x2)[q];
    float4 o;
    float* pa = &ba.x; float* pb = &bb.x; float* px = &xv.x; float* po = &o.x;
#pragma unroll
    for (int c = 0; c < 4; ++c) {
        float n = 4.0f * wh * winlen(j0 + c);
        po[c] = (pa[c] / n) * px[c] + pb[c] / n;
    }
    ((float4*)out)[q] = o;
}

extern "C" void kernel_launch(void* const* d_in, const int* in_sizes, int n_in,
                              void* d_out, int out_size, void* d_ws, size_t ws_size,
                              hipStream_t stream) {
    (void)in_sizes; (void)n_in; (void)out_size; (void)ws_size;
    const float* x = (const float*)d_in[0];
    const float* y = (const float*)d_in[1];
    float* out = (float*)d_out;
    float* ws  = (float*)d_ws;                 // 8 planes of 2048x2048 f32 (128 MB)

    const dim3 blk(256);
    const dim3 grdQ((unsigned)(PSIZE / 4 / 256));

    for (int b = 0; b < 2; ++b) {
        const float* xb = x + (size_t)b * 4 * PSIZE;
        const float* yb = y + (size_t)b * 4 * PSIZE;
        float* sums = ws;                      // planes 0..3 (later reused for A,bU)
        float* boxp = ws + 4 * PSIZE;          // planes 4..7

        reduce_kernel<<<grdQ, blk, 0, stream>>>(xb, yb, sums);
        box_kernel<<<dim3(32, 32, 4), blk, 0, stream>>>(sums, boxp);
        coeff_kernel<<<grdQ, blk, 0, stream>>>(boxp, sums, sums + PSIZE,
                                               out + 2 * PSIZE + (size_t)b * PSIZE);
        box_kernel<<<dim3(32, 32, 2), blk, 0, stream>>>(sums, boxp);
        final_kernel<<<grdQ, blk, 0, stream>>>(boxp, xb + 2 * PSIZE,
                                               out + (size_t)b * PSIZE);
    }
}